// DistanceDecoder_84963043049853
// MI455X (gfx1250) — compile-verified
//
#include <hip/hip_runtime.h>
#include <hip/hip_bf16.h>

// Problem constants (match reference)
#define BATCH   1024
#define PCA     32
#define ODIM    6144
#define NOBJ    20
#define WAVES   8
#define NT      256          // output columns per block: 8 waves * 2 subtiles * 16
#define MAX_TILES 84         // sum_o ceil(count_o/16) <= 64 + 19 < 84

typedef __attribute__((ext_vector_type(2))) float v2f;
typedef __attribute__((ext_vector_type(8))) float v8f;

// ---- kernel 1: zero per-object counters (graph-safe, no memset) ----
__global__ void dd_zero_counts(int* __restrict__ counts) {
    if (threadIdx.x < NOBJ) counts[threadIdx.x] = 0;
}

// ---- kernel 2: bucket batch rows by object label ----
__global__ void dd_build_lists(const int* __restrict__ labels,
                               int* __restrict__ counts,
                               int* __restrict__ lists) {
    int b = blockIdx.x * blockDim.x + threadIdx.x;
    if (b < BATCH) {
        int o = labels[b];
        int slot = atomicAdd(&counts[o], 1);
        lists[o * BATCH + slot] = b;
    }
}

// ---- kernel 3: compact (object, mtile) work list ----
__global__ void dd_plan_tiles(const int* __restrict__ counts,
                              int* __restrict__ tile_obj,
                              int* __restrict__ tile_m,
                              int* __restrict__ ntiles) {
    if (threadIdx.x == 0 && blockIdx.x == 0) {
        int t = 0;
        for (int o = 0; o < NOBJ; ++o) {
            int mt = (counts[o] + 15) >> 4;
            for (int m = 0; m < mt && t < MAX_TILES; ++m) {
                tile_obj[t] = o; tile_m[t] = m; ++t;
            }
        }
        *ntiles = t;
    }
}

// ---- kernel 4: grouped WMMA GEMM ----
// Block tile: 16 gathered rows x 256 cols. Each wave owns two independent
// 16x16 subtiles (two accumulators -> no WMMA->WMMA RAW chain stalls,
// 2x B-reuse of each A fragment). K = 32 = 8 x v_wmma_f32_16x16x4_f32.
__global__ __launch_bounds__(WAVES * 32)
void dd_pca_wmma(const float* __restrict__ latent,
                 const float* __restrict__ means,
                 const float* __restrict__ comp,
                 const int*   __restrict__ counts,
                 const int*   __restrict__ lists,
                 const int*   __restrict__ tile_obj,
                 const int*   __restrict__ tile_m,
                 const int*   __restrict__ ntiles,
                 float* __restrict__ out) {
    __shared__ float Alds[16 * 33];   // stride 33: bank-conflict-free column reads
    __shared__ int   idx_lds[16];

    const int t = blockIdx.y;
    if (t >= *ntiles) return;                       // uniform exit
    const int o     = tile_obj[t];
    const int mtile = tile_m[t];
    const int count = counts[o];

    const int tid = threadIdx.x;

    // stage gathered row indices for this 16-row M-tile (-1 marks padding)
    if (tid < 16) {
        int row = mtile * 16 + tid;
        idx_lds[tid] = (row < count) ? lists[o * BATCH + row] : -1;
    }
    __syncthreads();

    // stage gathered latent rows (16 x 32 f32) into LDS; pad rows -> 0
    for (int e = tid; e < 16 * PCA; e += WAVES * 32) {
        int r = e >> 5, c0 = e & 31;
        int bi = idx_lds[r];
        Alds[r * 33 + c0] = (bi >= 0) ? latent[bi * PCA + c0] : 0.0f;
    }
    __syncthreads();

    const int lane = tid & 31;
    const int wave = tid >> 5;
    const int half = lane >> 4;                     // 0: K pair {0,1}; 1: {2,3}
    const int r    = lane & 15;                     // A-matrix row M held by lane
    const int col0 = blockIdx.x * NT + wave * 32 + (lane & 15);
    const int col1 = col0 + 16;
    const float* Bc0 = comp + (size_t)o * PCA * ODIM + col0;

    v8f acc0 = {};
    v8f acc1 = {};
    #pragma unroll
    for (int kk = 0; kk < 8; ++kk) {                // K = 32 = 8 x 4
        const int kb = kk * 4 + half * 2;
        v2f a, b0, b1;
        // A 16x4 f32: lane holds (M=r, K=kb) and (M=r, K=kb+1)
        a.x = Alds[r * 33 + kb];
        a.y = Alds[r * 33 + kb + 1];
        // B 4x16 f32: lane holds (K=kb, N=col) and (K=kb+1, N=col)
        const float* bp = Bc0 + (size_t)kb * ODIM;
        b0.x = bp[0];
        b0.y = bp[ODIM];
        b1.x = bp[16];
        b1.y = bp[ODIM + 16];
        acc0 = __builtin_amdgcn_wmma_f32_16x16x4_f32(
            false, a, false, b0, (short)0, acc0, false, false);
        acc1 = __builtin_amdgcn_wmma_f32_16x16x4_f32(
            false, a, false, b1, (short)0, acc1, false, false);
    }

    // epilogue: preload row indices (no LDS reads in divergent path),
    // add mean, stream results to HBM with non-temporal hint.
    const float m0 = means[(size_t)o * ODIM + col0];
    const float m1 = means[(size_t)o * ODIM + col1];
    const int rbase = half * 8;                     // C VGPR j -> M = j + 8*half
    int bidx[8];
    #pragma unroll
    for (int j = 0; j < 8; ++j) bidx[j] = idx_lds[rbase + j];
    #pragma unroll
    for (int j = 0; j < 8; ++j) {
        if (bidx[j] >= 0) {
            float* op = out + (size_t)bidx[j] * ODIM;
            __builtin_nontemporal_store(acc0[j] + m0, op + col0);
            __builtin_nontemporal_store(acc1[j] + m1, op + col1);
        }
    }
}

extern "C" void kernel_launch(void* const* d_in, const int* in_sizes, int n_in,
                              void* d_out, int out_size, void* d_ws, size_t ws_size,
                              hipStream_t stream) {
    const float* latent = (const float*)d_in[0];   // [1024, 32]
    const int*   labels = (const int*)d_in[1];     // [1024]
    const float* means  = (const float*)d_in[2];   // [20, 6144]
    const float* comp   = (const float*)d_in[3];   // [20, 32, 6144]
    float* out = (float*)d_out;                    // [1024, 6144]

    // workspace layout (~83 KB): counts | lists | tile_obj | tile_m | ntiles
    char* ws = (char*)d_ws;
    int* counts   = (int*)ws;                                   // 20 ints
    int* lists    = (int*)(ws + 128);                           // 20*1024 ints
    int* tile_obj = (int*)(ws + 128 + NOBJ * BATCH * 4);        // 84 ints
    int* tile_m   = tile_obj + MAX_TILES;                       // 84 ints
    int* ntiles   = tile_m + MAX_TILES;                         // 1 int

    dd_zero_counts<<<1, 32, 0, stream>>>(counts);
    dd_build_lists<<<BATCH / 256, 256, 0, stream>>>(labels, counts, lists);
    dd_plan_tiles<<<1, 1, 0, stream>>>(counts, tile_obj, tile_m, ntiles);

    dim3 grid(ODIM / NT, MAX_TILES, 1);
    dd_pca_wmma<<<grid, WAVES * 32, 0, stream>>>(
        latent, means, comp, counts, lists, tile_obj, tile_m, ntiles, out);
}